// TransformerLayer_48773648613836
// MI455X (gfx1250) — compile-verified
//
#include <hip/hip_runtime.h>
#include <math.h>

// ---------------------------------------------------------------------------
// Shapes
// ---------------------------------------------------------------------------
#define BS 4
#define NN 256      // nodes
#define DX 256
#define DE 64
#define DY 64
#define NROW_X (BS*NN)             // 1024
#define NROW_E (BS*NN*NN)          // 262144
#define EBIG   ((size_t)NROW_E*DE) // 16,777,216 floats (E-sized)

typedef float fx2 __attribute__((ext_vector_type(2)));
typedef float fx8 __attribute__((ext_vector_type(8)));

// ---------------------------------------------------------------------------
// Generic GEMM: out[M,N] = act(A[M,K] @ W[K,N] + bias[N]) * mask(row)
// One wave computes a 16x32 output tile: two V_WMMA_F32_16X16X4_F32
// accumulators sharing each A fragment (K in steps of 4).
// K and N are compile-time so B addressing is immediate-offset; BOUNDS only
// for the tiny M=4 y-projections.
// ISA layout (§7.12.2):
//   A 16x4 : lane<16 -> M=lane, K={k0,k0+1}; lane>=16 -> M=lane-16, K={k0+2,k0+3}
//   B 4x16 : VGPR r, lane<16 -> K=k0+r, N=lane; lane>=16 -> K=k0+2+r, N=lane-16
//   C/D    : VGPR r, lane<16 -> M=r, N=lane;   lane>=16 -> M=r+8,  N=lane-16
// MASKMODE: 0 none, 1 node rows (m = b*256+n), 2 edge rows (m = b*65536+i*256+j)
// ---------------------------------------------------------------------------
template <int K, int N, int MASKMODE, bool RELU, bool BOUNDS>
__global__ __launch_bounds__(32)
void gemm16(const float* __restrict__ A, const float* __restrict__ W,
            const float* __restrict__ bias, float* __restrict__ out,
            int M, const int* __restrict__ mask)
{
    const int lane = threadIdx.x;
    const int half = lane >> 4;       // 0 or 1
    const int l16  = lane & 15;
    const int n0   = blockIdx.x * 32;
    const int m0   = blockIdx.y * 16;

    const int  mA  = m0 + l16;
    const bool mok = BOUNDS ? (mA < M) : true;
    const float* __restrict__ Arow = A + (size_t)(mok ? mA : 0) * K;
    const float* __restrict__ Wc0  = W + n0 + l16;
    const float* __restrict__ Wc1  = Wc0 + 16;

    fx8 acc0 = {}, acc1 = {};
    #pragma unroll 4
    for (int k0 = 0; k0 < K; k0 += 4) {
        if (!BOUNDS && (k0 + 64 < K)) __builtin_prefetch(Arow + k0 + 64, 0, 3);
        const int ka = k0 + half * 2;
        fx2 afrag, b0, b1;
        if (BOUNDS) {
            afrag.x = mok ? Arow[ka]     : 0.0f;
            afrag.y = mok ? Arow[ka + 1] : 0.0f;
        } else {
            afrag = *(const fx2*)(Arow + ka);       // global_load_b64
        }
        b0.x = Wc0[ka * N];       b0.y = Wc0[(ka + 1) * N];
        b1.x = Wc1[ka * N];       b1.y = Wc1[(ka + 1) * N];
        acc0 = __builtin_amdgcn_wmma_f32_16x16x4_f32(false, afrag, false, b0,
                                                     (short)0, acc0, false, false);
        acc1 = __builtin_amdgcn_wmma_f32_16x16x4_f32(false, afrag, false, b1,
                                                     (short)0, acc1, false, false);
    }

    const int   n   = n0 + l16;
    const float bn0 = bias[n];
    const float bn1 = bias[n + 16];
    #pragma unroll
    for (int r = 0; r < 8; ++r) {
        const int m = m0 + r + half * 8;
        if (BOUNDS && m >= M) continue;
        float f = 1.0f;
        if (MASKMODE == 1) {
            const int b = m >> 8, node = m & 255;
            f = (float)mask[(b << 8) + node];
        } else if (MASKMODE == 2) {
            const int b = m >> 16, i = (m >> 8) & 255, j = m & 255;
            f = (float)(mask[(b << 8) + i] * mask[(b << 8) + j]);
        }
        float v0 = acc0[r] + bn0;
        float v1 = acc1[r] + bn1;
        if (RELU) { v0 = fmaxf(v0, 0.0f); v1 = fmaxf(v1, 0.0f); }
        out[(size_t)m * N + n]      = v0 * f;
        out[(size_t)m * N + n + 16] = v1 * f;
    }
}

// ---------------------------------------------------------------------------
// Scores + E-FiLM: Y = (Q_i*K_j)/sqrt(32) * (E1+1) + E2
//                  EF = ye_add + (ye_mul+1)*Y      (input to e_out GEMM)
// grid = 262144 blocks (b,i,j), 256 threads (h*32+f)
// ---------------------------------------------------------------------------
__global__ void scores_kernel(const float* __restrict__ Q, const float* __restrict__ Kt,
                              const float* __restrict__ E1, const float* __restrict__ E2,
                              const float* __restrict__ yea, const float* __restrict__ yem,
                              float* __restrict__ Y, float* __restrict__ EF)
{
    const int t   = threadIdx.x;
    const int row = blockIdx.x;                 // b*65536 + i*256 + j
    const int b = row >> 16, i = (row >> 8) & 255, j = row & 255;
    const size_t idx = (size_t)row * 256 + t;
    const float q = Q[((size_t)(b << 8) + i) * 256 + t];
    const float k = Kt[((size_t)(b << 8) + j) * 256 + t];
    float yv = q * k * 0.17677669529663687f;    // 1/sqrt(32)
    yv = yv * (E1[idx] + 1.0f) + E2[idx];
    Y[idx]  = yv;
    EF[idx] = yea[(b << 8) + t] + (yem[(b << 8) + t] + 1.0f) * yv;
}

// ---------------------------------------------------------------------------
// Masked per-feature online softmax over j fused with weighted-V sum.
// grid = 1024 blocks (b,i), 256 threads (h*32+f)
// ---------------------------------------------------------------------------
__global__ void attn_kernel(const float* __restrict__ Y, const float* __restrict__ Vb,
                            const int* __restrict__ mask, float* __restrict__ wV)
{
    const int t  = threadIdx.x;
    const int bi = blockIdx.x;                  // b*256 + i
    const int b  = bi >> 8;
    float m = -3.4e38f, l = 0.0f, acc = 0.0f;
    for (int j = 0; j < NN; ++j) {
        if (mask[(b << 8) + j] != 0) {          // uniform branch across the block
            const float yv = Y[((size_t)bi * 256 + j) * 256 + t];
            const float nm = fmaxf(m, yv);
            const float sc = expf(m - nm);
            const float p  = expf(yv - nm);
            l   = l * sc + p;
            acc = acc * sc + p * Vb[((size_t)(b << 8) + j) * 256 + t];
            m = nm;
        }
    }
    wV[(size_t)bi * 256 + t] = acc / l;
}

// ---------------------------------------------------------------------------
// X-FiLM: newX_in = yx_add + (yx_mul+1)*wV
// ---------------------------------------------------------------------------
__global__ void film_x_kernel(const float* __restrict__ wv, const float* __restrict__ yxa,
                              const float* __restrict__ yxm, float* __restrict__ o)
{
    const int bi = blockIdx.x, t = threadIdx.x, b = bi >> 8;
    const size_t idx = (size_t)bi * 256 + t;
    o[idx] = yxa[(b << 8) + t] + (yxm[(b << 8) + t] + 1.0f) * wv[idx];
}

// ---------------------------------------------------------------------------
// LayerNorm: out = (x+res - mean)/sqrt(var+eps)*g + b  (population var)
// grid = rows, block = D threads
// ---------------------------------------------------------------------------
template <int D>
__global__ void ln_kernel(const float* __restrict__ x, const float* __restrict__ res,
                          const float* __restrict__ g, const float* __restrict__ bta,
                          float* __restrict__ out)
{
    __shared__ float red[D];
    const size_t row = blockIdx.x;
    const int d = threadIdx.x;
    float v = x[row * D + d] + res[row * D + d];
    red[d] = v; __syncthreads();
    for (int s = D / 2; s > 0; s >>= 1) { if (d < s) red[d] += red[d + s]; __syncthreads(); }
    const float mean = red[0] / (float)D; __syncthreads();
    const float dv = v - mean;
    red[d] = dv * dv; __syncthreads();
    for (int s = D / 2; s > 0; s >>= 1) { if (d < s) red[d] += red[d + s]; __syncthreads(); }
    const float var = red[0] / (float)D;
    out[row * D + d] = dv * rsqrtf(var + 1e-5f) * g[d] + bta[d];
}

// ---------------------------------------------------------------------------
// X feature aggregation: xf[b] = [mean, min, max, std(ddof=1)] over n, per d
// grid = BS*DX blocks, 256 threads (n)
// ---------------------------------------------------------------------------
__global__ void agg_x_kernel(const float* __restrict__ X, float* __restrict__ xf)
{
    __shared__ float ssum[256], ssq[256], smn[256], smx[256];
    const int b = blockIdx.x >> 8, d = blockIdx.x & 255, n = threadIdx.x;
    const float v = X[((size_t)((b << 8) + n)) * 256 + d];
    ssum[n] = v; ssq[n] = v * v; smn[n] = v; smx[n] = v; __syncthreads();
    for (int s = 128; s > 0; s >>= 1) {
        if (n < s) {
            ssum[n] += ssum[n + s]; ssq[n] += ssq[n + s];
            smn[n] = fminf(smn[n], smn[n + s]); smx[n] = fmaxf(smx[n], smx[n + s]);
        }
        __syncthreads();
    }
    if (n == 0) {
        const float sum = ssum[0];
        const float var = (ssq[0] - sum * sum / 256.0f) / 255.0f;
        xf[b * 1024 + d]       = sum / 256.0f;
        xf[b * 1024 + 256 + d] = smn[0];
        xf[b * 1024 + 512 + d] = smx[0];
        xf[b * 1024 + 768 + d] = sqrtf(fmaxf(var, 0.0f));
    }
}

// ---------------------------------------------------------------------------
// E feature aggregation over (i,j): 65536 elements per (b,d)
// grid = BS*DE blocks, 256 threads, each thread strides 256 iterations
// ---------------------------------------------------------------------------
__global__ void agg_e_kernel(const float* __restrict__ E, float* __restrict__ ef)
{
    __shared__ float ssum[256], ssq[256], smn[256], smx[256];
    const int b = blockIdx.x >> 6, d = blockIdx.x & 63, t = threadIdx.x;
    float sum = 0.0f, sq = 0.0f, mn = 3.4e38f, mx = -3.4e38f;
    const size_t base = (size_t)b * 65536 * 64;
    for (int it = 0; it < 256; ++it) {
        const float v = E[base + (size_t)(it * 256 + t) * 64 + d];
        sum += v; sq += v * v; mn = fminf(mn, v); mx = fmaxf(mx, v);
    }
    ssum[t] = sum; ssq[t] = sq; smn[t] = mn; smx[t] = mx; __syncthreads();
    for (int s = 128; s > 0; s >>= 1) {
        if (t < s) {
            ssum[t] += ssum[t + s]; ssq[t] += ssq[t + s];
            smn[t] = fminf(smn[t], smn[t + s]); smx[t] = fmaxf(smx[t], smx[t + s]);
        }
        __syncthreads();
    }
    if (t == 0) {
        const float cnt = 65536.0f;
        const float var = (ssq[0] - ssum[0] * ssum[0] / cnt) / (cnt - 1.0f);
        ef[b * 256 + d]       = ssum[0] / cnt;
        ef[b * 256 + 64 + d]  = smn[0];
        ef[b * 256 + 128 + d] = smx[0];
        ef[b * 256 + 192 + d] = sqrtf(fmaxf(var, 0.0f));
    }
}

// ---------------------------------------------------------------------------
// y path (tiny: 4x64): new_y MLP, residual LN, FFN, LN
// grid = BS blocks, 64 threads
// ---------------------------------------------------------------------------
__device__ inline float blk_sum64(float v, float* red, int d)
{
    red[d] = v; __syncthreads();
    for (int s = 32; s > 0; s >>= 1) { if (d < s) red[d] += red[d + s]; __syncthreads(); }
    const float r = red[0]; __syncthreads();
    return r;
}

__global__ void y_path_kernel(const float* __restrict__ y,
                              const float* __restrict__ xf, const float* __restrict__ ef,
                              const float* __restrict__ wyy, const float* __restrict__ byy,
                              const float* __restrict__ wxy, const float* __restrict__ bxy,
                              const float* __restrict__ wey, const float* __restrict__ bey,
                              const float* __restrict__ wo1, const float* __restrict__ bo1,
                              const float* __restrict__ wo2, const float* __restrict__ bo2,
                              const float* __restrict__ g1,  const float* __restrict__ bb1,
                              const float* __restrict__ wl1, const float* __restrict__ bl1,
                              const float* __restrict__ wl2, const float* __restrict__ bl2,
                              const float* __restrict__ g2,  const float* __restrict__ bb2,
                              float* __restrict__ out)
{
    __shared__ float sa[64], sb[64], red[64];
    const int b = blockIdx.x, d = threadIdx.x;

    float acc = byy[d];
    for (int k = 0; k < 64; ++k)   acc += y[b * 64 + k]   * wyy[k * 64 + d];
    acc += bxy[d];
    for (int k = 0; k < 1024; ++k) acc += xf[b * 1024 + k] * wxy[k * 64 + d];
    acc += bey[d];
    for (int k = 0; k < 256; ++k)  acc += ef[b * 256 + k]  * wey[k * 64 + d];
    sa[d] = acc; __syncthreads();

    float h = bo1[d];
    for (int k = 0; k < 64; ++k) h += sa[k] * wo1[k * 64 + d];
    h = fmaxf(h, 0.0f);
    sb[d] = h; __syncthreads();
    float ny = bo2[d];
    for (int k = 0; k < 64; ++k) ny += sb[k] * wo2[k * 64 + d];

    // yr = LN(y + new_y)
    float xv = y[b * 64 + d] + ny;
    float mean = blk_sum64(xv, red, d) / 64.0f;
    float dv = xv - mean;
    float var = blk_sum64(dv * dv, red, d) / 64.0f;
    const float yr = dv * rsqrtf(var + 1e-5f) * g1[d] + bb1[d];
    sa[d] = yr; __syncthreads();

    float h2 = bl1[d];
    for (int k = 0; k < 64; ++k) h2 += sa[k] * wl1[k * 64 + d];
    h2 = fmaxf(h2, 0.0f);
    sb[d] = h2; __syncthreads();
    float f = bl2[d];
    for (int k = 0; k < 64; ++k) f += sb[k] * wl2[k * 64 + d];

    // yo = LN(yr + f)
    xv = yr + f;
    mean = blk_sum64(xv, red, d) / 64.0f;
    dv = xv - mean;
    var = blk_sum64(dv * dv, red, d) / 64.0f;
    out[b * 64 + d] = dv * rsqrtf(var + 1e-5f) * g2[d] + bb2[d];
}

// ---------------------------------------------------------------------------
// Host launcher
// ---------------------------------------------------------------------------
extern "C" void kernel_launch(void* const* d_in, const int* in_sizes, int n_in,
                              void* d_out, int out_size, void* d_ws, size_t ws_size,
                              hipStream_t stream)
{
    const float* X    = (const float*)d_in[0];
    const float* E    = (const float*)d_in[1];
    const float* y    = (const float*)d_in[2];
    const int*   mask = (const int*)  d_in[3];
#define P(i) ((const float*)d_in[i])
    // Param leaf order: jax pytree flatten (sorted dict keys; 'b' < 'w', 'b' < 'g')
    // 4/5 e_add.b/w   6/7 e_mul.b/w   8/9 e_out.b/w   10/11 e_y.b/w
    // 12/13 k.b/w     14/15 linE1     16/17 linE2     18/19 linX1   20/21 linX2
    // 22/23 lin_y1    24/25 lin_y2    26/27 normE1.b/g 28/29 normE2.b/g
    // 30/31 normX1.b/g 32/33 normX2.b/g 34/35 norm_y1.b/g 36/37 norm_y2.b/g
    // 38/39 q.b/w     40/41 v.b/w     42/43 x_out     44/45 x_y
    // 46/47 y_e_add   48/49 y_e_mul   50/51 y_out1    52/53 y_out2
    // 54/55 y_x_add   56/57 y_x_mul   58/59 y_y

    float* ws = (float*)d_ws;
    float* bufA = ws;                 // E-sized
    float* bufB = ws + EBIG;          // E-sized
    float* bufC = ws + 2 * EBIG;      // E-sized
    float* sm   = ws + 3 * EBIG;      // small region
    float* Qb     = sm;               sm += NROW_X * DX;
    float* Kb     = sm;               sm += NROW_X * DX;
    float* Vb     = sm;               sm += NROW_X * DX;
    float* wV     = sm;               sm += NROW_X * DX;
    float* newXin = sm;               sm += NROW_X * DX;
    float* newX   = sm;               sm += NROW_X * DX;
    float* Xr     = sm;               sm += NROW_X * DX;
    float* hX     = sm;               sm += NROW_X * DX;
    float* fX     = sm;               sm += NROW_X * DX;
    float* yea    = sm;               sm += BS * DX;
    float* yem    = sm;               sm += BS * DX;
    float* yxa    = sm;               sm += BS * DX;
    float* yxm    = sm;               sm += BS * DX;
    float* xf     = sm;               sm += BS * 4 * DX;
    float* ef     = sm;               sm += BS * 4 * DE;

    float* Xo = (float*)d_out;                 // [4,256,256]
    float* Eo = Xo + (size_t)NROW_X * DX;      // [4,256,256,64]
    float* yo = Eo + EBIG;                     // [4,64]

    const dim3 b32(32);
    const dim3 gX (DX / 32, NROW_X / 16);      // (8, 64)   M=1024, N=256
    const dim3 gE (DX / 32, NROW_E / 16);      // (8,16384) M=262144, N=256
    const dim3 gEo(DE / 32, NROW_E / 16);      // (2,16384) M=262144, N=64
    const dim3 gY (DX / 32, 1);                // (8,1)     M=4, N=256

    // --- projections ---
    gemm16<DX, DX, 1, false, false><<<gX, b32, 0, stream>>>(X, P(39), P(38), Qb, NROW_X, mask); // Q
    gemm16<DX, DX, 1, false, false><<<gX, b32, 0, stream>>>(X, P(13), P(12), Kb, NROW_X, mask); // K
    gemm16<DX, DX, 1, false, false><<<gX, b32, 0, stream>>>(X, P(41), P(40), Vb, NROW_X, mask); // V
    gemm16<DE, DX, 2, false, false><<<gE, b32, 0, stream>>>(E, P(7),  P(6),  bufA, NROW_E, mask); // E1=e_mul
    gemm16<DE, DX, 2, false, false><<<gE, b32, 0, stream>>>(E, P(5),  P(4),  bufB, NROW_E, mask); // E2=e_add
    gemm16<DY, DX, 0, false, true ><<<gY, b32, 0, stream>>>(y, P(47), P(46), yea, BS, mask);    // y_e_add
    gemm16<DY, DX, 0, false, true ><<<gY, b32, 0, stream>>>(y, P(49), P(48), yem, BS, mask);    // y_e_mul
    gemm16<DY, DX, 0, false, true ><<<gY, b32, 0, stream>>>(y, P(55), P(54), yxa, BS, mask);    // y_x_add
    gemm16<DY, DX, 0, false, true ><<<gY, b32, 0, stream>>>(y, P(57), P(56), yxm, BS, mask);    // y_x_mul

    // --- attention scores + E-FiLM (EF overwrites E2 in bufB) ---
    scores_kernel<<<NROW_E, 256, 0, stream>>>(Qb, Kb, bufA, bufB, yea, yem, bufC, bufB);

    // --- softmax + weighted V (reads Y=bufC before it is recycled) ---
    attn_kernel<<<NROW_X, 256, 0, stream>>>(bufC, Vb, mask, wV);

    // --- E output path ---
    gemm16<DX, DE, 2, false, false><<<gEo, b32, 0, stream>>>(bufB, P(9), P(8), bufA, NROW_E, mask); // newE=e_out(EF)
    ln_kernel<DE><<<NROW_E, DE, 0, stream>>>(E, bufA, P(27), P(26), bufB);                          // Er=LN(E+newE)
    gemm16<DE, DE, 0, true,  false><<<gEo, b32, 0, stream>>>(bufB, P(15), P(14), bufC, NROW_E, mask); // hE=relu(linE1)
    gemm16<DE, DE, 0, false, false><<<gEo, b32, 0, stream>>>(bufC, P(17), P(16), bufA, NROW_E, mask); // fE=linE2
    ln_kernel<DE><<<NROW_E, DE, 0, stream>>>(bufB, bufA, P(29), P(28), Eo);                         // Eo=LN(Er+fE)

    // --- X output path ---
    film_x_kernel<<<NROW_X, 256, 0, stream>>>(wV, yxa, yxm, newXin);
    gemm16<DX, DX, 1, false, false><<<gX, b32, 0, stream>>>(newXin, P(43), P(42), newX, NROW_X, mask); // x_out
    ln_kernel<DX><<<NROW_X, DX, 0, stream>>>(X, newX, P(31), P(30), Xr);                            // Xr
    gemm16<DX, DX, 0, true,  false><<<gX, b32, 0, stream>>>(Xr, P(19), P(18), hX, NROW_X, mask);    // relu(linX1)
    gemm16<DX, DX, 0, false, false><<<gX, b32, 0, stream>>>(hX, P(21), P(20), fX, NROW_X, mask);    // linX2
    ln_kernel<DX><<<NROW_X, DX, 0, stream>>>(Xr, fX, P(33), P(32), Xo);                             // Xo

    // --- y path ---
    agg_x_kernel<<<BS * DX, 256, 0, stream>>>(X, xf);
    agg_e_kernel<<<BS * DE, 256, 0, stream>>>(E, ef);
    y_path_kernel<<<BS, DY, 0, stream>>>(y, xf, ef,
        P(59), P(58), P(45), P(44), P(11), P(10),
        P(51), P(50), P(53), P(52),
        P(35), P(34), P(23), P(22), P(25), P(24), P(37), P(36), yo);
#undef P
}